// SetAbstraction_5016521802585
// MI455X (gfx1250) — compile-verified
//
#include <hip/hip_runtime.h>

#define B_  8
#define N_  2048
#define K_  32
#define C1  64
#define C2  128
#define C3  256

typedef __attribute__((ext_vector_type(2))) float v2f;
typedef __attribute__((ext_vector_type(8))) float v8f;

// ---------------------------------------------------------------------------
// Stage 1: kNN (squared distance, top-32 per query, stable tie-break on index)
// One block = 256 queries of one batch; whole batch cloud staged in LDS.
// ---------------------------------------------------------------------------
__global__ __launch_bounds__(256) void knn_kernel(const float* __restrict__ pts,
                                                  int* __restrict__ knn) {
    __shared__ float4 sp[N_];
    const int b     = blockIdx.x / (N_ / 256);
    const int qbase = (blockIdx.x % (N_ / 256)) * 256;
    const float* pb = pts + (size_t)b * N_ * 3;

    for (int j = threadIdx.x; j < N_; j += 256) {
        sp[j] = make_float4(pb[j * 3 + 0], pb[j * 3 + 1], pb[j * 3 + 2], 0.f);
    }
    __syncthreads();

    const int q = qbase + threadIdx.x;
    const float4 p = sp[q];

    float bd[K_];
    int   bi[K_];
#pragma unroll
    for (int s = 0; s < K_; ++s) {
        float4 c = sp[s];
        float dx = p.x - c.x, dy = p.y - c.y, dz = p.z - c.z;
        bd[s] = dx * dx + dy * dy + dz * dz;
        bi[s] = s;
    }
    // current worst of the 32 (larger dist, tie -> larger index loses)
    float wd = bd[0]; int wi = bi[0]; int wslot = 0;
#pragma unroll
    for (int s = 1; s < K_; ++s) {
        bool worse = (bd[s] > wd) || (bd[s] == wd && bi[s] > wi);
        if (worse) { wd = bd[s]; wi = bi[s]; wslot = s; }
    }

    for (int j = K_; j < N_; ++j) {
        float4 c = sp[j];
        float dx = p.x - c.x, dy = p.y - c.y, dz = p.z - c.z;
        float d  = dx * dx + dy * dy + dz * dz;
        // strict '<' is correct: on a tie the newer (larger) index loses,
        // matching stable ascending argsort.
        if (d < wd) {
#pragma unroll
            for (int s = 0; s < K_; ++s)
                if (s == wslot) { bd[s] = d; bi[s] = j; }
            wd = -1.f; wi = -1;
#pragma unroll
            for (int s = 0; s < K_; ++s) {
                bool worse = (bd[s] > wd) || (bd[s] == wd && bi[s] > wi);
                if (worse) { wd = bd[s]; wi = bi[s]; wslot = s; }
            }
        }
    }

#pragma unroll
    for (int s = 0; s < K_; ++s)
        knn[((size_t)b * N_ + q) * K_ + s] = bi[s];
}

// ---------------------------------------------------------------------------
// Stage 2: per-point MLP 3->64->128->256 (ReLU), fp32 WMMA 16x16x4.
// One wave owns a 16-point row tile; activations round-trip through a
// bank-conflict-free LDS slice to re-shape C/D layout into A-fragment layout.
// ---------------------------------------------------------------------------
#define MLP_WAVES 4
#define S1 68    // 64 + 4 pad  (stride mod 64 banks = 4 -> conflict free)
#define S2 132   // 128 + 4 pad

__global__ __launch_bounds__(128) void mlp_kernel(
        const float* __restrict__ pts,
        const float* __restrict__ W1, const float* __restrict__ b1,
        const float* __restrict__ W2, const float* __restrict__ b2,
        const float* __restrict__ W3, const float* __restrict__ b3,
        float* __restrict__ feat) {
    __shared__ float act1[MLP_WAVES][16 * S1];
    __shared__ float act2[MLP_WAVES][16 * S2];

    const int wave = threadIdx.x >> 5;
    const int lane = threadIdx.x & 31;
    const int m    = lane & 15;
    const int half = lane >> 4;
    const int p0   = (blockIdx.x * MLP_WAVES + wave) * 16;

    float* A1 = act1[wave];
    float* A2 = act2[wave];

    // ---- layer 1 : X[16x3 padded to 16x4] * W1[4x64] ----
    v2f a;
    {
        const int row = p0 + m;
        const int k0  = 2 * half;          // 0 or 2, both valid coords
        a.x = pts[(size_t)row * 3 + k0];
        a.y = (half == 0) ? pts[(size_t)row * 3 + 1] : 0.f;  // k=1 valid, k=3 pad
    }
#pragma unroll
    for (int t = 0; t < C1 / 16; ++t) {
        const int col = t * 16 + m;
        v2f bf;
        bf.x = W1[(2 * half) * C1 + col];                   // K row 0 or 2
        bf.y = (half == 0) ? W1[1 * C1 + col] : 0.f;        // K row 1 or 3(pad)
        v8f c = {};
        c = __builtin_amdgcn_wmma_f32_16x16x4_f32(false, a, false, bf,
                                                  (short)0, c, false, false);
        const float bias = b1[col];
#pragma unroll
        for (int r = 0; r < 8; ++r) {
            const int rr = r + 8 * half;
            float v = c[r] + bias;
            A1[rr * S1 + col] = v > 0.f ? v : 0.f;
        }
    }

    // ---- layer 2 : A1[16x64] * W2[64x128] ----
#pragma unroll
    for (int t = 0; t < C2 / 16; ++t) {
        const int col = t * 16 + m;
        v8f c = {};
        for (int ks = 0; ks < C1 / 4; ++ks) {
            const int kb = ks * 4 + 2 * half;
            v2f aa, bb;
            aa.x = A1[m * S1 + kb];
            aa.y = A1[m * S1 + kb + 1];
            bb.x = W2[(size_t)kb * C2 + col];
            bb.y = W2[(size_t)(kb + 1) * C2 + col];
            c = __builtin_amdgcn_wmma_f32_16x16x4_f32(false, aa, false, bb,
                                                      (short)0, c, false, false);
        }
        const float bias = b2[col];
#pragma unroll
        for (int r = 0; r < 8; ++r) {
            const int rr = r + 8 * half;
            float v = c[r] + bias;
            A2[rr * S2 + col] = v > 0.f ? v : 0.f;
        }
    }

    // ---- layer 3 : A2[16x128] * W3[128x256] -> feature table ----
    for (int t = 0; t < C3 / 16; ++t) {
        const int col = t * 16 + m;
        v8f c = {};
        for (int ks = 0; ks < C2 / 4; ++ks) {
            const int kb = ks * 4 + 2 * half;
            v2f aa, bb;
            aa.x = A2[m * S2 + kb];
            aa.y = A2[m * S2 + kb + 1];
            bb.x = W3[(size_t)kb * C3 + col];
            bb.y = W3[(size_t)(kb + 1) * C3 + col];
            c = __builtin_amdgcn_wmma_f32_16x16x4_f32(false, aa, false, bb,
                                                      (short)0, c, false, false);
        }
        const float bias = b3[col];
#pragma unroll
        for (int r = 0; r < 8; ++r) {
            const int rr = r + 8 * half;
            float v = c[r] + bias;
            feat[(size_t)(p0 + rr) * C3 + col] = v > 0.f ? v : 0.f;
        }
    }
}

// ---------------------------------------------------------------------------
// Stage 3: gather 32 neighbor features (L2-resident table) and mean-pool.
// One block = one query; 64 threads x float4 cover 256 channels.
// ---------------------------------------------------------------------------
__global__ __launch_bounds__(64) void pool_kernel(const int* __restrict__ knn,
                                                  const float* __restrict__ feat,
                                                  float* __restrict__ out) {
    __shared__ int sidx[K_];
    const size_t q = blockIdx.x;          // 0 .. B*N-1
    const int    b = (int)(q / N_);
    if (threadIdx.x < K_) sidx[threadIdx.x] = knn[q * K_ + threadIdx.x];
    __syncthreads();

    const int c0 = threadIdx.x * 4;
    float4 acc = make_float4(0.f, 0.f, 0.f, 0.f);
    for (int k = 0; k < K_; ++k) {
        const float4 v = *(const float4*)(feat + ((size_t)b * N_ + sidx[k]) * C3 + c0);
        acc.x += v.x; acc.y += v.y; acc.z += v.z; acc.w += v.w;
    }
    const float s = 1.0f / (float)K_;
    *(float4*)(out + q * C3 + c0) =
        make_float4(acc.x * s, acc.y * s, acc.z * s, acc.w * s);
}

// ---------------------------------------------------------------------------
extern "C" void kernel_launch(void* const* d_in, const int* in_sizes, int n_in,
                              void* d_out, int out_size, void* d_ws, size_t ws_size,
                              hipStream_t stream) {
    const float* pts = (const float*)d_in[0];
    const float* W1  = (const float*)d_in[1];
    const float* b1  = (const float*)d_in[2];
    const float* W2  = (const float*)d_in[3];
    const float* b2  = (const float*)d_in[4];
    const float* W3  = (const float*)d_in[5];
    const float* b3  = (const float*)d_in[6];

    int*   knn  = (int*)d_ws;                                   // 2 MB
    float* feat = (float*)((char*)d_ws +
                           (size_t)B_ * N_ * K_ * sizeof(int)); // 16 MB

    knn_kernel<<<B_ * (N_ / 256), 256, 0, stream>>>(pts, knn);
    mlp_kernel<<<(B_ * N_ / 16) / MLP_WAVES, 128, 0, stream>>>(
        pts, W1, b1, W2, b2, W3, b3, feat);
    pool_kernel<<<B_ * N_, 64, 0, stream>>>(knn, feat, (float*)d_out);
}